// WeightedGraphConv_61495341744683
// MI455X (gfx1250) — compile-verified
//
#include <hip/hip_runtime.h>
#include <hip/hip_bf16.h>

// ---------------------------------------------------------------------------
// WeightedGraphConv: out = segment_sum( x[src] * (ew/deg[dst]) , dst ) @ W + b
// N=40000 nodes, E=640000 edges, D_IN=D_OUT=128, all fp32.
// Stage 1: zero h (=d_out) and deg (=d_ws)
// Stage 2: deg[dst] += 1           (scalar f32 atomics, L2-resident)
// Stage 3: h[dst]  += w * x[src]   (wave-per-edge, float4 per lane, f32 atomics)
// Stage 4: out = h @ W + b         (V_WMMA_F32_16X16X4_F32, fp32 end-to-end)
// ---------------------------------------------------------------------------

typedef float v2f __attribute__((ext_vector_type(2)));
typedef float v8f __attribute__((ext_vector_type(8)));

#define DFEAT 128

__global__ void wgc_zero_kernel(float* __restrict__ p, int n) {
    int i = blockIdx.x * blockDim.x + threadIdx.x;
    if (i < n) p[i] = 0.0f;
}

__global__ void wgc_deg_kernel(const int* __restrict__ dst,
                               float* __restrict__ deg, int E) {
    int i = blockIdx.x * blockDim.x + threadIdx.x;
    if (i < E) atomicAdd(&deg[dst[i]], 1.0f);
}

// One wave32 per edge: lane l handles feature dims [4l, 4l+4).
__global__ void wgc_scatter_kernel(const float* __restrict__ x,
                                   const int* __restrict__ src,
                                   const int* __restrict__ dst,
                                   const float* __restrict__ ew,
                                   const float* __restrict__ deg,
                                   float* __restrict__ h, int E) {
    int gtid = blockIdx.x * blockDim.x + threadIdx.x;
    int e    = gtid >> 5;           // global wave id == edge id
    int lane = threadIdx.x & 31;
    if (e >= E) return;
    int s = src[e];
    int d = dst[e];
    float dg = deg[d];
    float w  = ew[e] * (dg > 0.0f ? 1.0f / dg : 0.0f);
    const float4 v = ((const float4*)(x + (size_t)s * DFEAT))[lane];
    float* hd = h + (size_t)d * DFEAT + lane * 4;
    atomicAdd(hd + 0, w * v.x);
    atomicAdd(hd + 1, w * v.y);
    atomicAdd(hd + 2, w * v.z);
    atomicAdd(hd + 3, w * v.w);
}

// In-place GEMM: `out` holds H on entry; each block owns 16 rows, stages them
// through LDS, then overwrites with H@W + b. 256 threads = 8 waves; wave w
// computes the 16x16 tile at columns [16w, 16w+16) via 32x WMMA f32 16x16x4.
#define LDS_STRIDE 132   // 128 + 4 pad: row stride 4 banks -> conflict-free A reads

__global__ void wgc_gemm_kernel(const float* __restrict__ W,
                                const float* __restrict__ bias,
                                float* __restrict__ out) {
    __shared__ float Hs[16 * LDS_STRIDE];

    const int tid  = threadIdx.x;          // 0..255
    const int m0   = blockIdx.x * 16;

    // Stage the 16x128 H tile into LDS (coalesced global reads).
    const float* gsrc = out + (size_t)m0 * DFEAT;
    for (int i = tid; i < 16 * DFEAT; i += 256) {
        int row = i >> 7, col = i & 127;
        Hs[row * LDS_STRIDE + col] = gsrc[i];
    }
    __syncthreads();

    const int wave = tid >> 5;             // 0..7 -> N tile
    const int lane = tid & 31;
    const int mrow = lane & 15;            // A: M index; B/C: N index within tile
    const int koff = (lane >> 4) * 2;      // 0 for lanes 0-15, 2 for lanes 16-31
    const int n0   = wave * 16;

    const float* hrow = Hs + mrow * LDS_STRIDE;
    const float* wcol = W + (n0 + mrow);

    v8f c = {};
    #pragma unroll
    for (int k = 0; k < DFEAT; k += 4) {
        // A fragment (16x4 f32): lane holds H[mrow][k+koff], H[mrow][k+koff+1]
        v2f a = *(const v2f*)(hrow + k + koff);
        // B fragment (4x16 f32): lane holds W[k+koff][n0+mrow], W[k+koff+1][...]
        v2f bf;
        bf.x = wcol[(size_t)(k + koff) * DFEAT];
        bf.y = wcol[(size_t)(k + koff + 1) * DFEAT];
        // 8 args: (neg_a, A, neg_b, B, c_mod, C, reuse_a, reuse_b)
        c = __builtin_amdgcn_wmma_f32_16x16x4_f32(
                false, a, false, bf, (short)0, c, false, false);
    }

    // C/D layout: VGPR r -> (m = r + 8*(lane>>4), n = lane&15)
    const int n     = n0 + mrow;
    const int mbase = m0 + ((lane >> 4) << 3);
    const float bv  = bias[n];
    #pragma unroll
    for (int r = 0; r < 8; ++r) {
        out[(size_t)(mbase + r) * DFEAT + n] = c[r] + bv;
    }
}

extern "C" void kernel_launch(void* const* d_in, const int* in_sizes, int n_in,
                              void* d_out, int out_size, void* d_ws, size_t ws_size,
                              hipStream_t stream) {
    const float* x   = (const float*)d_in[0];
    const int*   src = (const int*)d_in[1];
    const int*   dst = (const int*)d_in[2];
    const float* ew  = (const float*)d_in[3];
    const float* W   = (const float*)d_in[4];
    const float* b   = (const float*)d_in[5];
    (void)n_in; (void)ws_size;

    const int N = in_sizes[0] / DFEAT;     // 40000
    const int E = in_sizes[1];             // 640000

    float* h   = (float*)d_out;            // accumulate H in-place in d_out
    float* deg = (float*)d_ws;             // N floats of scratch

    // 1) zero h and deg
    {
        int n = N * DFEAT;
        wgc_zero_kernel<<<(n + 255) / 256, 256, 0, stream>>>(h, n);
        wgc_zero_kernel<<<(N + 255) / 256, 256, 0, stream>>>(deg, N);
    }
    // 2) in-degree
    wgc_deg_kernel<<<(E + 255) / 256, 256, 0, stream>>>(dst, deg, E);
    // 3) weighted scatter: one wave per edge, 8 edges per 256-thread block
    {
        int waves  = E;
        int blocks = (waves + 7) / 8;
        wgc_scatter_kernel<<<blocks, 256, 0, stream>>>(x, src, dst, ew, deg, h, E);
    }
    // 4) in-place WMMA GEMM: out = H @ W + b
    wgc_gemm_kernel<<<N / 16, 256, 0, stream>>>(W, b, h);

    (void)out_size;
}